// VDPDropout_56092272885821
// MI455X (gfx1250) — compile-verified
//
#include <hip/hip_runtime.h>
#include <hip/hip_bf16.h>
#include <stdint.h>

// Problem constants (from reference): B=32, H=2048, drop=0.2
#define BB 32
#define HH 2048
#define SCALE 1.25f           // 1/(1-0.2)
#define ROWS_PER_BLOCK 16     // Sigma rows handled per block
#define TPB 256               // 8 wave32s

typedef float v4f __attribute__((ext_vector_type(4)));

// ---------------------------------------------------------------------------
// Kernel A: mu_out = keep ? mu*scale : 0 ;  f = (mu_out != 0) ? scale : 0
// 65536 elements total — negligible cost, sets up the row-factor vector.
// ---------------------------------------------------------------------------
__global__ __launch_bounds__(TPB)
void vdp_mu_kernel(const float* __restrict__ mu_in,
                   const int* __restrict__ keep_mask,
                   float* __restrict__ mu_out,
                   float* __restrict__ f) {
    const int idx = blockIdx.x * TPB + threadIdx.x;   // grid sized exactly B*H
    const float m  = mu_in[idx];
    const int   k  = keep_mask[idx];
    const float mo = k ? (m * SCALE) : 0.0f;
    mu_out[idx] = mo;
    // matches reference: nz = (mu_out != 0), not just the keep mask
    f[idx] = (mo != 0.0f) ? SCALE : 0.0f;
}

// ---------------------------------------------------------------------------
// Kernel B: Sigma_out[b,i,j] = Sigma_in[b,i,j] * f[b,i] * f[b,j]
// Pure streaming: 512 MiB in + 512 MiB out @ 23.3 TB/s  ->  ~46 us floor.
//  - f[b,0:H] (8 KB) staged into LDS via async global->LDS B128 DMA,
//    completion via ASYNCcnt, then reused across 16 rows per block.
//  - Sigma traffic uses non-temporal B128 loads/stores (1 GB >> 192 MB L2).
// ---------------------------------------------------------------------------
__global__ __launch_bounds__(TPB)
void vdp_sigma_kernel(const float* __restrict__ Sigma_in,
                      const float* __restrict__ f,
                      float* __restrict__ Sigma_out) {
    __shared__ float fsh[HH];                       // 8 KB of 320 KB/WGP

    const int b     = blockIdx.x >> 7;              // / (HH/ROWS_PER_BLOCK = 128)
    const int itile = blockIdx.x & 127;
    const int i0    = itile * ROWS_PER_BLOCK;
    const int tid   = threadIdx.x;

    // ---- Stage f[b][0:2048] into LDS with async tensor-path copies ----
    // 512 x b128 transfers, 2 per thread. Each wave waits its own ASYNCcnt,
    // then the workgroup barrier makes the LDS contents visible to all.
    {
        const float* gsrc0 = f + (size_t)b * HH + tid * 4;
        const float* gsrc1 = gsrc0 + 1024;
        uint32_t l0 = (uint32_t)(uintptr_t)(&fsh[tid * 4]);   // LDS byte offset
        uint32_t l1 = l0 + 1024 * 4;
        asm volatile("global_load_async_to_lds_b128 %0, %1, off"
                     :: "v"(l0), "v"(gsrc0) : "memory");
        asm volatile("global_load_async_to_lds_b128 %0, %1, off"
                     :: "v"(l1), "v"(gsrc1) : "memory");
        asm volatile("s_wait_asynccnt 0" ::: "memory");
    }
    __syncthreads();

    // Per-thread j-lane factors are loop-invariant: keep them in VGPRs.
    const int j0 = tid * 4;                          // 16B-aligned float4 lane
    const v4f fj0 = *(const v4f*)(&fsh[j0]);
    const v4f fj1 = *(const v4f*)(&fsh[j0 + 1024]);

    const size_t base = ((size_t)b * HH + i0) * (size_t)HH;
    const float* __restrict__ src = Sigma_in + base;
    float* __restrict__ dst = Sigma_out + base;

#pragma unroll 4
    for (int r = 0; r < ROWS_PER_BLOCK; ++r) {
        const float fi = fsh[i0 + r];
        const size_t ro = (size_t)r * HH;

        v4f a0 = __builtin_nontemporal_load((const v4f*)(src + ro + j0));
        v4f a1 = __builtin_nontemporal_load((const v4f*)(src + ro + j0 + 1024));

        v4f o0 = a0 * fj0 * fi;                      // f_i*f_j = 1.5625 or 0
        v4f o1 = a1 * fj1 * fi;

        __builtin_nontemporal_store(o0, (v4f*)(dst + ro + j0));
        __builtin_nontemporal_store(o1, (v4f*)(dst + ro + j0 + 1024));
    }
}

// ---------------------------------------------------------------------------
extern "C" void kernel_launch(void* const* d_in, const int* in_sizes, int n_in,
                              void* d_out, int out_size, void* d_ws, size_t ws_size,
                              hipStream_t stream) {
    const float* mu_in    = (const float*)d_in[0];
    const float* Sigma_in = (const float*)d_in[1];
    const int*   keep     = (const int*)d_in[2];

    float* mu_out    = (float*)d_out;                 // [B,H]
    float* Sigma_out = (float*)d_out + (size_t)BB * HH; // [B,H,H], concatenated
    float* f         = (float*)d_ws;                  // [B,H] row factors (256 KB)

    // A: compute mu_out and the 0/scale factor vector.
    vdp_mu_kernel<<<(BB * HH) / TPB, TPB, 0, stream>>>(mu_in, keep, mu_out, f);

    // B: stream Sigma. 32*(2048/16) = 4096 blocks, 256 threads each.
    vdp_sigma_kernel<<<BB * (HH / ROWS_PER_BLOCK), TPB, 0, stream>>>(
        Sigma_in, f, Sigma_out);
}